// Geo3DLoss_46643344835317
// MI455X (gfx1250) — compile-verified
//
#include <hip/hip_runtime.h>

typedef __attribute__((ext_vector_type(2))) float v2f;
typedef __attribute__((ext_vector_type(8))) float v8f;

#define BATCHES 32768
#define NPTS    256
#define WAVES_PER_BLOCK 8

__device__ __forceinline__ float wave_sum32(float v) {
    v += __shfl_xor(v, 16, 32);
    v += __shfl_xor(v, 8, 32);
    v += __shfl_xor(v, 4, 32);
    v += __shfl_xor(v, 2, 32);
    v += __shfl_xor(v, 1, 32);
    return v;
}

__global__ __launch_bounds__(256) void geo3d_batch_kernel(
        const float* __restrict__ kp, const int* __restrict__ perm,
        float* __restrict__ partial) {
    // per-wave LDS slice: src[256*3] then dst[256*3]
    __shared__ float lds[WAVES_PER_BLOCK * 2 * NPTS * 3];
    __shared__ float blk_red[WAVES_PER_BLOCK];

    const int lane = threadIdx.x & 31;
    const int wave = threadIdx.x >> 5;
    const int b    = blockIdx.x * WAVES_PER_BLOCK + wave;
    float* srcS = &lds[wave * 2 * NPTS * 3];
    float* dstS = srcS + NPTS * 3;

    const int pb = perm[b];
    const float* __restrict__ src = kp + (size_t)b  * (NPTS * 3);
    const float* __restrict__ dst = kp + (size_t)pb * (NPTS * 3);

    // ---- load 8 points/lane, stage to LDS, accumulate mean partials ----
    float s[8][3], d[8][3];
    float sm0 = 0.f, sm1 = 0.f, sm2 = 0.f, dm0 = 0.f, dm1 = 0.f, dm2 = 0.f;
#pragma unroll
    for (int t = 0; t < 8; ++t) {
        const int n = lane + 32 * t;
        const float* sp = src + n * 3;
        const float* dp = dst + n * 3;
        s[t][0] = sp[0]; s[t][1] = sp[1]; s[t][2] = sp[2];
        d[t][0] = dp[0]; d[t][1] = dp[1]; d[t][2] = dp[2];
        srcS[n * 3 + 0] = s[t][0]; srcS[n * 3 + 1] = s[t][1]; srcS[n * 3 + 2] = s[t][2];
        dstS[n * 3 + 0] = d[t][0]; dstS[n * 3 + 1] = d[t][1]; dstS[n * 3 + 2] = d[t][2];
        sm0 += s[t][0]; sm1 += s[t][1]; sm2 += s[t][2];
        dm0 += d[t][0]; dm1 += d[t][1]; dm2 += d[t][2];
    }
    __syncthreads();

    const float inv_n = 1.0f / (float)NPTS;
    sm0 = wave_sum32(sm0) * inv_n; sm1 = wave_sum32(sm1) * inv_n; sm2 = wave_sum32(sm2) * inv_n;
    dm0 = wave_sum32(dm0) * inv_n; dm1 = wave_sum32(dm1) * inv_n; dm2 = wave_sum32(dm2) * inv_n;

    // ---- unnormalized cross-covariance C[i][j] = sum_n dst_d[i]*src_d[j], and var ----
    float C[9] = {0.f,0.f,0.f,0.f,0.f,0.f,0.f,0.f,0.f};
    float var = 0.f;
#pragma unroll
    for (int t = 0; t < 8; ++t) {
        const float sd0 = s[t][0] - sm0, sd1 = s[t][1] - sm1, sd2 = s[t][2] - sm2;
        const float dd0 = d[t][0] - dm0, dd1 = d[t][1] - dm1, dd2 = d[t][2] - dm2;
        C[0] += dd0 * sd0; C[1] += dd0 * sd1; C[2] += dd0 * sd2;
        C[3] += dd1 * sd0; C[4] += dd1 * sd1; C[5] += dd1 * sd2;
        C[6] += dd2 * sd0; C[7] += dd2 * sd1; C[8] += dd2 * sd2;
        var  += sd0 * sd0 + sd1 * sd1 + sd2 * sd2;
    }
#pragma unroll
    for (int i = 0; i < 9; ++i) C[i] = wave_sum32(C[i]);
    var = wave_sum32(var);

    // ---- R = U@Vt = polar factor of C via det-scaled Newton iteration ----
    float X[9];
#pragma unroll
    for (int i = 0; i < 9; ++i) X[i] = C[i];
#pragma unroll
    for (int it = 0; it < 8; ++it) {
        const float c00 = X[4]*X[8] - X[5]*X[7];
        const float c01 = X[5]*X[6] - X[3]*X[8];
        const float c02 = X[3]*X[7] - X[4]*X[6];
        const float c10 = X[2]*X[7] - X[1]*X[8];
        const float c11 = X[0]*X[8] - X[2]*X[6];
        const float c12 = X[1]*X[6] - X[0]*X[7];
        const float c20 = X[1]*X[5] - X[2]*X[4];
        const float c21 = X[2]*X[3] - X[0]*X[5];
        const float c22 = X[0]*X[4] - X[1]*X[3];
        float det = X[0]*c00 + X[1]*c01 + X[2]*c02;
        const float ad = fmaxf(fabsf(det), 1e-30f);
        det = (det >= 0.f) ? fmaxf(det, 1e-30f) : fminf(det, -1e-30f);
        const float g  = 1.0f / cbrtf(ad);        // |det|^(-1/3)
        const float ga = 0.5f * g;                // scales X
        const float gb = 0.5f / (g * det);        // scales cof(X) = det * X^{-T}
        X[0] = ga*X[0] + gb*c00;  X[1] = ga*X[1] + gb*c01;  X[2] = ga*X[2] + gb*c02;
        X[3] = ga*X[3] + gb*c10;  X[4] = ga*X[4] + gb*c11;  X[5] = ga*X[5] + gb*c12;
        X[6] = ga*X[6] + gb*c20;  X[7] = ga*X[7] + gb*c21;  X[8] = ga*X[8] + gb*c22;
    }

    // sum(S) = trace(R^T C); both C and var unnormalized by N -> ratio unchanged
    float sumS = 0.f;
#pragma unroll
    for (int i = 0; i < 9; ++i) sumS += X[i] * C[i];
    const float scl = sumS / var;
    const float T0 = dm0 - scl * (X[0]*sm0 + X[1]*sm1 + X[2]*sm2);
    const float T1 = dm1 - scl * (X[3]*sm0 + X[4]*sm1 + X[5]*sm2);
    const float T2 = dm2 - scl * (X[6]*sm0 + X[7]*sm1 + X[8]*sm2);

    // ---- transform via V_WMMA_F32_16X16X4_F32 ----
    // Homogeneous trick: A = [src.x src.y src.z 1] (16x4), B[k][j] = scale*R[j][k] for
    // k<3, B[3][j] = T[j]. Then D[m][j] = trans[m][j] directly. Columns j>=3 all zero.
    // f32 operand layout: lanes 0-15 hold K=0 (v0) / K=1 (v1); lanes 16-31 K=2 (v0) / K=3 (v1).
    const int   j       = lane & 15;
    const bool  hi      = lane >= 16;
    const float hiMask  = hi ? 1.0f : 0.0f;
    const float loMask  = 1.0f - hiMask;
    const float colMask = (j < 3) ? 1.0f : 0.0f;
    const int   jc      = (j < 3) ? j : 0;       // clamped column for safe LDS reads

    const float r0 = (j == 0) ? X[0] : ((j == 1) ? X[3] : ((j == 2) ? X[6] : 0.f));
    const float r1 = (j == 0) ? X[1] : ((j == 1) ? X[4] : ((j == 2) ? X[7] : 0.f));
    const float r2 = (j == 0) ? X[2] : ((j == 1) ? X[5] : ((j == 2) ? X[8] : 0.f));
    const float Tj = (j == 0) ? T0 : ((j == 1) ? T1 : ((j == 2) ? T2 : 0.f));

    v2f bmat;
    bmat.x = hi ? (scl * r2) : (scl * r0);   // rows K=2 / K=0
    bmat.y = hi ? Tj         : (scl * r1);   // rows K=3 / K=1

    const v8f zero = {};
    float loss = 0.f;
#pragma unroll
    for (int tile = 0; tile < 16; ++tile) {
        const int m = tile * 16 + (lane & 15);
        // unconditional LDS loads; branch-free operand selection
        const float ax_lo = srcS[m * 3 + (hi ? 2 : 0)];  // x (lo) or z (hi): address select only
        const float ay_ld = srcS[m * 3 + 1];             // y, loaded by all lanes
        v2f amat;
        amat.x = ax_lo;
        amat.y = fmaf(loMask, ay_ld, hiMask);            // exactly y (lo) or exactly 1.0 (hi)

        v8f acc = __builtin_amdgcn_wmma_f32_16x16x4_f32(
            false, amat, false, bmat, (short)0, zero, false, false);

        // D layout: VGPR v holds (M = v + 8*hi, N = lane&15); masked accumulation
#pragma unroll
        for (int v = 0; v < 8; ++v) {
            const int p = tile * 16 + v + (hi ? 8 : 0);
            const float dv = dstS[p * 3 + jc];           // unconditional load
            loss += colMask * fabsf(acc[v] - dv);
        }
    }

    loss = wave_sum32(loss);
    if (lane == 0) blk_red[wave] = loss;
    __syncthreads();
    if (threadIdx.x == 0) {
        float bl = 0.f;
#pragma unroll
        for (int w = 0; w < WAVES_PER_BLOCK; ++w) bl += blk_red[w];
        partial[blockIdx.x] = bl;
    }
}

__global__ __launch_bounds__(256) void geo3d_reduce_kernel(
        const float* __restrict__ partial, float* __restrict__ out) {
    __shared__ float red[256];
    float acc = 0.f;
    for (int i = threadIdx.x; i < BATCHES / WAVES_PER_BLOCK; i += 256) acc += partial[i];
    red[threadIdx.x] = acc;
    __syncthreads();
    for (int off = 128; off > 0; off >>= 1) {
        if ((int)threadIdx.x < off) red[threadIdx.x] += red[threadIdx.x + off];
        __syncthreads();
    }
    if (threadIdx.x == 0)
        out[0] = red[0] * (1.0f / ((float)BATCHES * (float)NPTS * 3.0f));
}

extern "C" void kernel_launch(void* const* d_in, const int* in_sizes, int n_in,
                              void* d_out, int out_size, void* d_ws, size_t ws_size,
                              hipStream_t stream) {
    (void)in_sizes; (void)n_in; (void)out_size; (void)ws_size;
    const float* kp   = (const float*)d_in[0];
    const int*   perm = (const int*)d_in[1];
    float* out = (float*)d_out;
    float* ws  = (float*)d_ws;   // 4096 per-block partials (16 KB)

    geo3d_batch_kernel<<<BATCHES / WAVES_PER_BLOCK, 256, 0, stream>>>(kp, perm, ws);
    geo3d_reduce_kernel<<<1, 256, 0, stream>>>(ws, out);
}